// PostProcessPose_35931696398817
// MI455X (gfx1250) — compile-verified
//
#include <hip/hip_runtime.h>
#include <stdint.h>

// ---- problem constants (fixed by reference setup_inputs) -------------------
#define B_   4
#define Q_   100
#define K_   17
#define HH   128
#define WW   128
#define QSLICES 4                 // q-slice blocks for occupancy
#define QPS   (Q_ / QSLICES)      // 25 boxes per slice
#define HUMAN_CONF 0.7f
#define KP_CONF    0.5f
#define NMS_THR    0.5f
#define NEG_INF   (-__builtin_inff())

// ---- CDNA5 async global->LDS path (probe-confirmed present) ----------------
#define AS1 __attribute__((address_space(1)))
#define AS3 __attribute__((address_space(3)))

typedef int v4i_ __attribute__((vector_size(16)));   // b128 payload type

#if defined(__has_builtin)
#  if __has_builtin(__builtin_amdgcn_global_load_async_to_lds_b128)
#    define HAVE_ASYNC_B128 1
#  endif
#  if __has_builtin(__builtin_amdgcn_s_wait_asynccnt)
#    define HAVE_WAIT_ASYNC 1
#  endif
#endif

__device__ __forceinline__ void async_copy_b128(const float* gsrc, float* ldst) {
#ifdef HAVE_ASYNC_B128
  // gfx1250: copies 16 bytes from the global address gsrc to the LDS address ldst, tracked by ASYNCcnt
  __builtin_amdgcn_global_load_async_to_lds_b128(
      (AS1 v4i_*)(uint64_t)gsrc,
      (AS3 v4i_*)(uint32_t)(uint64_t)ldst,
      /*offset=*/0, /*cpol=*/0);
#else
  *(float4*)ldst = *(const float4*)gsrc;  // synchronous fallback
#endif
}

__device__ __forceinline__ void wait_async_lds() {
#ifdef HAVE_ASYNC_B128
#  ifdef HAVE_WAIT_ASYNC
  __builtin_amdgcn_s_wait_asynccnt(0);
#  else
  asm volatile("s_wait_asynccnt 0" ::: "memory");
#  endif
#endif
}

__device__ __forceinline__ int imax_(int a, int b) { return a > b ? a : b; }
__device__ __forceinline__ int imin_(int a, int b) { return a < b ? a : b; }

__device__ __forceinline__ float iou_box(const float* a, const float* b) {
  float areaA = fmaxf(a[2] - a[0], 0.f) * fmaxf(a[3] - a[1], 0.f);
  float areaB = fmaxf(b[2] - b[0], 0.f) * fmaxf(b[3] - b[1], 0.f);
  float lx = fmaxf(a[0], b[0]), ly = fmaxf(a[1], b[1]);
  float rx = fminf(a[2], b[2]), ry = fminf(a[3], b[3]);
  float inter = fmaxf(rx - lx, 0.f) * fmaxf(ry - ly, 0.f);
  float uni = areaA + areaB - inter;
  return inter / fmaxf(uni, 1e-9f);
}

// ---------------------------------------------------------------------------
// Kernel A: per-batch softmax + box convert + rank sort + greedy NMS.
// One 128-thread block per batch (Q=100). Writes scores/boxes/keep outputs
// and a per-box descriptor {keep, x1,y1,x2,y2} (heatmap coords) to workspace.
// ---------------------------------------------------------------------------
__global__ void __launch_bounds__(128)
pose_nms_kernel(const float* __restrict__ logits,   // [B,Q,2]
                const float* __restrict__ pboxes,   // [B,Q,4] cxcywh in [0,1]
                const int* __restrict__ p_ih, const int* __restrict__ p_iw,
                float* __restrict__ out_scores,     // [B,Q]
                float* __restrict__ out_boxes,      // [B,Q,4]
                float* __restrict__ out_keep,       // [B,Q]
                int* __restrict__ ws)               // [B,Q,5]
{
  __shared__ float s_box[Q_][4];
  __shared__ float s_key[Q_];
  __shared__ float s_score[Q_];
  __shared__ int   s_valid[Q_];
  __shared__ int   s_order[Q_];
  __shared__ int   s_sup[Q_];
  __shared__ int   s_keep[Q_];

  const int b = blockIdx.x;
  const int t = threadIdx.x;
  const float img_w = (float)p_iw[0];
  const float img_h = (float)p_ih[0];

  if (t < Q_) {
    const float l0 = logits[(b * Q_ + t) * 2 + 0];
    const float l1 = logits[(b * Q_ + t) * 2 + 1];
    const float m  = fmaxf(l0, l1);
    const float e0 = __expf(l0 - m), e1 = __expf(l1 - m);
    const float score = fmaxf(e0, e1) / (e0 + e1);
    const int   label = (l1 > l0) ? 1 : 0;             // argmax, first-on-tie
    const int   valid = (label == 0) && (score >= HUMAN_CONF);

    const float cx = pboxes[(b * Q_ + t) * 4 + 0];
    const float cy = pboxes[(b * Q_ + t) * 4 + 1];
    const float w  = pboxes[(b * Q_ + t) * 4 + 2];
    const float h  = pboxes[(b * Q_ + t) * 4 + 3];
    s_box[t][0] = (cx - 0.5f * w) * img_w;
    s_box[t][1] = (cy - 0.5f * h) * img_h;
    s_box[t][2] = (cx + 0.5f * w) * img_w;
    s_box[t][3] = (cy + 0.5f * h) * img_h;
    s_score[t] = score;
    s_valid[t] = valid;
    s_key[t]   = valid ? score : NEG_INF;
    s_sup[t]   = 0;
  }
  __syncthreads();

  // rank = position in descending sort (stable-ascending-argsort reversed)
  if (t < Q_) {
    const float kq = s_key[t];
    int r = 0;
    for (int j = 0; j < Q_; ++j) {
      const float kj = s_key[j];
      r += (kj > kq) || ((kj == kq) && (j > t));
    }
    s_order[r] = t;
  }
  __syncthreads();

  // greedy NMS over sorted positions; IoU row computed in parallel
  for (int i = 0; i < Q_; ++i) {
    const int  qi = s_order[i];
    const bool ki = s_valid[qi] && (s_sup[i] == 0);   // block-uniform
    if (ki && t < Q_ && t != i) {
      const int qj = s_order[t];
      if (iou_box(s_box[qi], s_box[qj]) > NMS_THR) s_sup[t] = 1;
    }
    if (t == 0) s_keep[qi] = ki ? 1 : 0;
    __syncthreads();
  }

  // outputs + ROI descriptors
  if (t < Q_) {
    const int kp = s_keep[t];
    out_keep[b * Q_ + t]   = (float)kp;
    out_scores[b * Q_ + t] = s_score[t] * (float)kp;

    const int bi0 = (int)s_box[t][0];   // trunc-toward-zero == astype(int32)
    const int bi1 = (int)s_box[t][1];
    const int bi2 = (int)s_box[t][2];
    const int bi3 = (int)s_box[t][3];
    out_boxes[(b * Q_ + t) * 4 + 0] = (float)(bi0 * kp);
    out_boxes[(b * Q_ + t) * 4 + 1] = (float)(bi1 * kp);
    out_boxes[(b * Q_ + t) * 4 + 2] = (float)(bi2 * kp);
    out_boxes[(b * Q_ + t) * 4 + 3] = (float)(bi3 * kp);

    const float sx = (float)WW / img_w;
    const float sy = (float)HH / img_h;
    int* w5 = ws + (b * Q_ + t) * 5;
    w5[0] = kp;
    w5[1] = imax_((int)((float)bi0 * sx), 0);
    w5[2] = imax_((int)((float)bi1 * sy), 0);
    w5[3] = imin_((int)((float)bi2 * sx) + 1, WW);
    w5[4] = imin_((int)((float)bi3 * sy) + 1, HH);
  }
}

// ---------------------------------------------------------------------------
// Kernel B: one block per (b, k, q-slice). Stage the 64KB heatmap into LDS
// via gfx1250 async global->LDS b128 loads; then ONE WAVE PER BOX does a
// pure wave32 shuffle argmax from LDS — no intra-loop barriers at all.
// 256 threads = 8 waves; each wave walks q = slice_base + wave, step 8.
// ---------------------------------------------------------------------------
__global__ void __launch_bounds__(256)
pose_roi_kernel(const float* __restrict__ heat,   // [B,K,HH,WW]
                const int* __restrict__ ws,       // [B,Q,5]
                const int* __restrict__ p_ih, const int* __restrict__ p_iw,
                float* __restrict__ out_kps)      // [B,Q,K,3]
{
  __shared__ __align__(16) float s_hm[HH * WW];   // 64 KB (of 320 KB/WGP)

  const int k  = blockIdx.x;
  const int b  = blockIdx.y;
  const int qs = blockIdx.z;                      // q-slice
  const int t  = threadIdx.x;
  const int wave = t >> 5, lane = t & 31;
  const float* src = heat + (size_t)(b * K_ + k) * (HH * WW);

  // ---- async stage: 4096 x b128 transfers, 16 per thread ----
  #pragma unroll
  for (int it = 0; it < (HH * WW / 4) / 256; ++it) {
    const int v4 = it * 256 + t;
    async_copy_b128(src + v4 * 4, &s_hm[v4 * 4]);
  }
  wait_async_lds();      // s_wait_asynccnt 0
  __syncthreads();

  const float inv_sx = (float)p_iw[0] / (float)WW;  // 1/sx
  const float inv_sy = (float)p_ih[0] / (float)HH;  // 1/sy

  const int q_lo = qs * QPS;
  for (int q = q_lo + wave; q < q_lo + QPS; q += 8) {
    const int* w5 = ws + (b * Q_ + q) * 5;        // wave-uniform -> scalarized
    const int kp = w5[0], x1 = w5[1], y1 = w5[2], x2 = w5[3], y2 = w5[4];
    float* o = out_kps + ((size_t)(b * Q_ + q) * K_ + k) * 3;

    if (!(kp && (x2 > x1) && (y2 > y1))) {
      if (lane == 0) { o[0] = 0.f; o[1] = 0.f; o[2] = 0.f; }
      continue;
    }

    const int rw = x2 - x1;
    const int n  = (y2 - y1) * rw;
    float bv = NEG_INF;
    int   bi = 0x7fffffff;
    for (int j = lane; j < n; j += 32) {
      const int r = j / rw, c = j - r * rw;
      const int flat = (y1 + r) * WW + (x1 + c);
      const float v = s_hm[flat];
      if (v > bv || (v == bv && flat < bi)) { bv = v; bi = flat; }
    }
    // wave32 shuffle argmax, lowest-flat-index tie-break (== jnp argmax)
    #pragma unroll
    for (int off = 16; off > 0; off >>= 1) {
      const float ov = __shfl_down(bv, off);
      const int   oi = __shfl_down(bi, off);
      if (ov > bv || (ov == bv && oi < bi)) { bv = ov; bi = oi; }
    }
    if (lane == 0) {
      const float col = (float)(bi % WW);
      const float row = (float)(bi / WW);
      o[0] = col * inv_sx;
      o[1] = row * inv_sy;
      o[2] = (bv < KP_CONF) ? 0.0f : bv;
    }
  }
}

// ---------------------------------------------------------------------------
extern "C" void kernel_launch(void* const* d_in, const int* in_sizes, int n_in,
                              void* d_out, int out_size, void* d_ws, size_t ws_size,
                              hipStream_t stream) {
  const float* logits = (const float*)d_in[0];   // [4,100,2]
  const float* pboxes = (const float*)d_in[1];   // [4,100,4]
  const float* heat   = (const float*)d_in[2];   // [4,17,128,128]
  const int*   p_ih   = (const int*)d_in[3];     // scalar
  const int*   p_iw   = (const int*)d_in[4];     // scalar

  float* out        = (float*)d_out;
  float* out_scores = out;                                   // 400
  float* out_boxes  = out + B_ * Q_;                         // 1600
  float* out_kps    = out + B_ * Q_ * 5;                     // 20400
  float* out_keep   = out + B_ * Q_ * 5 + B_ * Q_ * K_ * 3;  // 400
  int*   ws         = (int*)d_ws;                            // [4,100,5] ints

  pose_nms_kernel<<<dim3(B_), dim3(128), 0, stream>>>(
      logits, pboxes, p_ih, p_iw, out_scores, out_boxes, out_keep, ws);

  pose_roi_kernel<<<dim3(K_, B_, QSLICES), dim3(256), 0, stream>>>(
      heat, ws, p_ih, p_iw, out_kps);
}